// EntropyBottleneck_16518444220824
// MI455X (gfx1250) — compile-verified
//
#include <hip/hip_runtime.h>
#include <hip/hip_bf16.h>
#include <math.h>
#include <stdint.h>

// EntropyBottleneck forward (noise-quantize + likelihood), MI455X / gfx1250.
// Memory-bound elementwise kernel: 512MB traffic floor ~22us @ 23.3TB/s.
// gfx1250 paths: v_tanh_f32 (builtin preferred, asm fallback), async
// global->LDS staging (ASYNCcnt), non-temporal streaming accesses.

#define CHN   64
#define PPC   44                    // folded params per channel
#define PTOT  (CHN * PPC)           // 2816 floats = 11264 bytes
#define PCHUNKS ((PTOT * 4) / 16)   // 704 x b128 async chunks

// ---------- fast device math (gfx1250 transcendentals) ----------
__device__ __forceinline__ float eb_tanh(float x) {
#if __has_builtin(__builtin_amdgcn_tanhf)
    // compiler schedules the TRANS hazard and interleaves independent tanhs
    return __builtin_amdgcn_tanhf(x);
#else
#warning "gfx1250 tanh builtin not available; using inline-asm v_tanh_f32"
    float r;
    asm("v_tanh_f32 %0, %1\n\t"
        "s_delay_alu instid0(TRANS32_DEP_1)"
        : "=v"(r) : "v"(x));
    return r;
#endif
}

__device__ __forceinline__ float eb_exp2(float x) {
#if __has_builtin(__builtin_amdgcn_exp2f)
    return __builtin_amdgcn_exp2f(x);        // v_exp_f32
#else
#warning "exp2 builtin not available; using inline-asm v_exp_f32"
    float r;
    asm("v_exp_f32 %0, %1\n\t"
        "s_delay_alu instid0(TRANS32_DEP_1)"
        : "=v"(r) : "v"(x));
    return r;
#endif
}

__device__ __forceinline__ float eb_log2(float x) {
#if __has_builtin(__builtin_amdgcn_logf)
    return __builtin_amdgcn_logf(x);         // v_log_f32
#else
#warning "log2 builtin not available; using inline-asm v_log_f32"
    float r;
    asm("v_log_f32 %0, %1\n\t"
        "s_delay_alu instid0(TRANS32_DEP_1)"
        : "=v"(r) : "v"(x));
    return r;
#endif
}

__device__ __forceinline__ float eb_rcp(float x) {
#if __has_builtin(__builtin_amdgcn_rcpf)
    return __builtin_amdgcn_rcpf(x);         // v_rcp_f32
#else
#warning "rcp builtin not available; using inline-asm v_rcp_f32"
    float r;
    asm("v_rcp_f32 %0, %1\n\t"
        "s_delay_alu instid0(TRANS32_DEP_1)"
        : "=v"(r) : "v"(x));
    return r;
#endif
}

__device__ __forceinline__ float eb_sigmoid(float z) {
    // 1/(1+e^-z) = 1/(1+2^(-z*log2e))
    float e = eb_exp2(-1.4426950408889634f * z);
    return eb_rcp(1.0f + e);
}

__device__ __forceinline__ float eb_softplus(float x) {
    // ln(1+e^x) = ln2 * log2(1 + 2^(x*log2e)); guard large x (2^y overflow)
    float e = eb_exp2(1.4426950408889634f * x);
    float l = 0.6931471805599453f * eb_log2(1.0f + e);
    return (x > 20.0f) ? x : l;
}

// ---------- prep: fold softplus(m)/b/tanh(f) into a [C][44] table ----------
__global__ void __launch_bounds__(64) eb_prep(
    const float* __restrict__ m0, const float* __restrict__ b0, const float* __restrict__ f0,
    const float* __restrict__ m1, const float* __restrict__ b1, const float* __restrict__ f1,
    const float* __restrict__ m2, const float* __restrict__ b2, const float* __restrict__ f2,
    const float* __restrict__ m3, const float* __restrict__ b3, const float* __restrict__ f3,
    float* __restrict__ P)
{
    int c = threadIdx.x;
    if (c >= CHN) return;
    float* o = P + c * PPC;
#pragma unroll
    for (int i = 0; i < 3; i++) {
        o[0 + i] = eb_softplus(m0[c * 3 + i]);   // W0 (3x1)
        o[3 + i] = b0[c * 3 + i];                // B0
        o[6 + i] = eb_tanh(f0[c * 3 + i]);       // T0
    }
#pragma unroll
    for (int k = 0; k < 9; k++) o[9 + k] = eb_softplus(m1[c * 9 + k]);   // W1 (3x3 row-major)
#pragma unroll
    for (int i = 0; i < 3; i++) { o[18 + i] = b1[c * 3 + i]; o[21 + i] = eb_tanh(f1[c * 3 + i]); }
#pragma unroll
    for (int k = 0; k < 9; k++) o[24 + k] = eb_softplus(m2[c * 9 + k]);  // W2 (3x3)
#pragma unroll
    for (int i = 0; i < 3; i++) { o[33 + i] = b2[c * 3 + i]; o[36 + i] = eb_tanh(f2[c * 3 + i]); }
#pragma unroll
    for (int j = 0; j < 3; j++) o[39 + j] = eb_softplus(m3[c * 3 + j]);  // W3 (1x3)
    o[42] = b3[c];
    o[43] = eb_tanh(f3[c]);
}

// ---------- the 4-layer gated chain, all params in registers ----------
__device__ __forceinline__ float eb_chain(float x, const float* __restrict__ p) {
    float v0 = fmaf(p[0], x, p[3]);
    float v1 = fmaf(p[1], x, p[4]);
    float v2 = fmaf(p[2], x, p[5]);
    v0 = fmaf(p[6], eb_tanh(v0), v0);
    v1 = fmaf(p[7], eb_tanh(v1), v1);
    v2 = fmaf(p[8], eb_tanh(v2), v2);
    float w0 = fmaf(p[9],  v0, fmaf(p[10], v1, fmaf(p[11], v2, p[18])));
    float w1 = fmaf(p[12], v0, fmaf(p[13], v1, fmaf(p[14], v2, p[19])));
    float w2 = fmaf(p[15], v0, fmaf(p[16], v1, fmaf(p[17], v2, p[20])));
    w0 = fmaf(p[21], eb_tanh(w0), w0);
    w1 = fmaf(p[22], eb_tanh(w1), w1);
    w2 = fmaf(p[23], eb_tanh(w2), w2);
    float u0 = fmaf(p[24], w0, fmaf(p[25], w1, fmaf(p[26], w2, p[33])));
    float u1 = fmaf(p[27], w0, fmaf(p[28], w1, fmaf(p[29], w2, p[34])));
    float u2 = fmaf(p[30], w0, fmaf(p[31], w1, fmaf(p[32], w2, p[35])));
    u0 = fmaf(p[36], eb_tanh(u0), u0);
    u1 = fmaf(p[37], eb_tanh(u1), u1);
    u2 = fmaf(p[38], eb_tanh(u2), u2);
    float y = fmaf(p[39], u0, fmaf(p[40], u1, fmaf(p[41], u2, p[42])));
    y = fmaf(p[43], eb_tanh(y), y);
    return y;
}

__device__ __forceinline__ void eb_element(
    int i, const float* __restrict__ in, const float* __restrict__ noise,
    float* __restrict__ out, float* __restrict__ lik, const float* __restrict__ pr)
{
    float a  = __builtin_nontemporal_load(in + i);
    float nz = __builtin_nontemporal_load(noise + i);
    float x  = a + nz;
    __builtin_nontemporal_store(x, out + i);

    float lo = eb_chain(x - 0.5f, pr);
    float hi = eb_chain(x + 0.5f, pr);
    float t  = lo + hi;
    float s  = (t > 0.0f) ? -1.0f : ((t < 0.0f) ? 1.0f : 0.0f);  // -sign(lo+hi)
    float l  = fabsf(eb_sigmoid(s * hi) - eb_sigmoid(s * lo));
    l = fmaxf(l, 1e-9f);                                          // low_bound
    __builtin_nontemporal_store(l, lik + i);
}

// ---------- main streaming kernel ----------
__global__ void __launch_bounds__(256) eb_main(
    const float* __restrict__ in, const float* __restrict__ noise,
    const float* __restrict__ P, float* __restrict__ out,
    float* __restrict__ lik, int total)
{
    __shared__ float sP[PTOT];

    // Stage the 11KB folded-parameter table into LDS via the CDNA5
    // async global->LDS path (ASYNCcnt tracked), then hoist to registers.
    {
        unsigned lbase = (unsigned)(uintptr_t)&sP[0];   // low 32 bits of flat LDS addr = LDS offset
        for (int k = threadIdx.x; k < PCHUNKS; k += 256) {
            unsigned loff = lbase + (unsigned)(k * 16);
            const float* g = P + k * 4;
            asm volatile("global_load_async_to_lds_b128 %0, %1, off"
                         :: "v"(loff), "v"(g) : "memory");
        }
#if __has_builtin(__builtin_amdgcn_s_wait_asynccnt)
        __builtin_amdgcn_s_wait_asynccnt(0);
#else
        asm volatile("s_wait_asynccnt 0x0" ::: "memory");
#endif
        __syncthreads();
    }

    int i      = blockIdx.x * 256 + threadIdx.x;
    int stride = gridDim.x * 256;                 // multiple of 64 -> channel loop-invariant
    int c      = i & (CHN - 1);

    float pr[PPC];
#pragma unroll
    for (int k = 0; k < PPC; k++) pr[k] = sP[c * PPC + k];   // 11 x ds_load_b128

    // unroll x2: batch the NT loads/stores for latency hiding
    for (; i + stride < total; i += 2 * stride) {
        eb_element(i,          in, noise, out, lik, pr);
        eb_element(i + stride, in, noise, out, lik, pr);
    }
    if (i < total) eb_element(i, in, noise, out, lik, pr);
}

extern "C" void kernel_launch(void* const* d_in, const int* in_sizes, int n_in,
                              void* d_out, int out_size, void* d_ws, size_t ws_size,
                              hipStream_t stream) {
    const float* in    = (const float*)d_in[0];
    const float* noise = (const float*)d_in[1];
    const float* m0 = (const float*)d_in[2],  *b0 = (const float*)d_in[3],  *f0 = (const float*)d_in[4];
    const float* m1 = (const float*)d_in[5],  *b1 = (const float*)d_in[6],  *f1 = (const float*)d_in[7];
    const float* m2 = (const float*)d_in[8],  *b2 = (const float*)d_in[9],  *f2 = (const float*)d_in[10];
    const float* m3 = (const float*)d_in[11], *b3 = (const float*)d_in[12], *f3 = (const float*)d_in[13];

    float* P = (float*)d_ws;                 // [CHN][PPC] folded params (written every call)
    int total = in_sizes[0];                 // N*C
    float* out = (float*)d_out;              // outputs (N,C)
    float* lik = out + total;                // likelihood (N,C)

    hipLaunchKernelGGL(eb_prep, dim3(1), dim3(64), 0, stream,
                       m0, b0, f0, m1, b1, f1, m2, b2, f2, m3, b3, f3, P);

    int blocks = (total + 255) / 256;
    if (blocks > 4096) blocks = 4096;        // grid-stride; stride stays a multiple of 64
    hipLaunchKernelGGL(eb_main, dim3(blocks), dim3(256), 0, stream,
                       in, noise, P, out, lik, total);
}